// ZeroMaskedLayerNorm_61598420959415
// MI455X (gfx1250) — compile-verified
//
#include <hip/hip_runtime.h>
#include <hip/hip_bf16.h>

typedef __attribute__((ext_vector_type(2))) float v2f;
typedef __attribute__((ext_vector_type(4))) float f4;
typedef __attribute__((ext_vector_type(8))) float v8f;

static constexpr int   kB      = 8;
static constexpr int   kC      = 32;
static constexpr int   kS      = 64 * 64 * 64;   // 262144 spatial / sample (2^18)
static constexpr int   kCH     = 4096;           // spatial chunk per block
static constexpr int   kNCHUNK = kS / kCH;       // 64
static constexpr int   kNBLK1  = kB * kNCHUNK;   // 512 reduction blocks
static constexpr int   kNBLK3  = 4096;           // normalize blocks
static constexpr float kEPS    = 1e-5f;

// ---------------------------------------------------------------------------
// Pass 1: per-(sample, chunk) masked sum / sumsq / count.
// Mask (channel 0) staged in LDS once, reused across all 32 channels.
// Sums accumulated with V_WMMA_F32_16X16X4_F32 (B = ones -> row sums).
// All hot loops have compile-time-constant trip counts so the compiler can
// fully unroll them: no EXEC bookkeeping, and 4 b128 loads in flight per wave.
// ---------------------------------------------------------------------------
__global__ __launch_bounds__(256) void zmln_reduce(const float* __restrict__ x,
                                                   float* __restrict__ partials)
{
    __shared__ f4       smask[kCH / 4];
    __shared__ unsigned scnt;
    __shared__ float    swsum[8];
    __shared__ float    swss[8];

    const int b     = blockIdx.x / kNCHUNK;
    const int chunk = blockIdx.x % kNCHUNK;
    const int tid   = threadIdx.x;
    const int wave  = tid >> 5;
    const int lane  = tid & 31;

    if (tid == 0) scnt = 0u;
    __syncthreads();

    const f4* xf4 = (const f4*)x;
    // channel-0 base of this chunk, in f4 units
    const size_t base0 = (size_t)b * kC * (kS / 4) + (size_t)chunk * (kCH / 4);

    // ---- Phase 0: build mask in LDS, count selected voxels -----------------
    // kCH/4 = 1024 f4 elements, 256 threads -> exactly 4 per thread.
    unsigned cnt = 0;
#pragma unroll
    for (int jj = 0; jj < (kCH / 4) / 256; ++jj) {
        const int j = tid + jj * 256;
        f4 c0 = xf4[base0 + j];
        f4 m;
        m.x = (c0.x != 0.0f) ? 1.0f : 0.0f;
        m.y = (c0.y != 0.0f) ? 1.0f : 0.0f;
        m.z = (c0.z != 0.0f) ? 1.0f : 0.0f;
        m.w = (c0.w != 0.0f) ? 1.0f : 0.0f;
        cnt += (unsigned)(m.x + m.y + m.z + m.w);
        smask[j] = m;
    }
    atomicAdd(&scnt, cnt);          // LDS integer atomic: order-independent
    __syncthreads();

    // ---- Phase 1: stream 32 channels, WMMA-accumulate masked sums ----------
    v8f acc_s0 = {}, acc_s1 = {}, acc_q0 = {}, acc_q1 = {};
    const v2f ones = {1.0f, 1.0f};

    for (int c = 0; c < kC; ++c) {
        const size_t cb = (size_t)(b * kC + c) * (kS / 4) + (size_t)chunk * (kCH / 4);
        if (c + 1 < kC)   // hint next channel's segment -> global_prefetch_b8
            __builtin_prefetch(xf4 + cb + (kS / 4) + (wave * 4) * 32 + lane, 0, 3);
        // kCH/128 = 32 wave-iterations, 8 waves -> exactly 4 per wave.
#pragma unroll
        for (int ii = 0; ii < (kCH / 128) / 8; ++ii) {
            const int i  = wave + ii * 8;
            const int fi = i * 32 + lane;
            f4 v  = xf4[cb + fi];
            f4 m  = smask[fi];
            f4 xm = v * m;          // masked x
            f4 xq = xm * v;         // masked x^2 (m*m == m)
            v2f a0 = {xm.x, xm.y};
            v2f a1 = {xm.z, xm.w};
            v2f q0 = {xq.x, xq.y};
            v2f q1 = {xq.z, xq.w};
            acc_s0 = __builtin_amdgcn_wmma_f32_16x16x4_f32(false, a0, false, ones,
                                                           (short)0, acc_s0, false, false);
            acc_s1 = __builtin_amdgcn_wmma_f32_16x16x4_f32(false, a1, false, ones,
                                                           (short)0, acc_s1, false, false);
            acc_q0 = __builtin_amdgcn_wmma_f32_16x16x4_f32(false, q0, false, ones,
                                                           (short)0, acc_q0, false, false);
            acc_q1 = __builtin_amdgcn_wmma_f32_16x16x4_f32(false, q1, false, ones,
                                                           (short)0, acc_q1, false, false);
        }
    }

    // ---- Fold: all 16 accumulator columns are identical --------------------
    v8f as = acc_s0 + acc_s1;
    v8f aq = acc_q0 + acc_q1;
    float ls = 0.0f, lq = 0.0f;
#pragma unroll
    for (int k = 0; k < 8; ++k) { ls += as[k]; lq += aq[k]; }
    // lane n holds rows 0..7 of column n, lane n+16 holds rows 8..15
    ls += __shfl_xor(ls, 16, 32);
    lq += __shfl_xor(lq, 16, 32);

    if (lane == 0) { swsum[wave] = ls; swss[wave] = lq; }
    __syncthreads();

    if (tid == 0) {
        float s = 0.0f, q = 0.0f;
#pragma unroll
        for (int w = 0; w < 8; ++w) { s += swsum[w]; q += swss[w]; }
        partials[(size_t)blockIdx.x * 3 + 0] = s;
        partials[(size_t)blockIdx.x * 3 + 1] = q;
        partials[(size_t)blockIdx.x * 3 + 2] = (float)(scnt * kC);
    }
}

// ---------------------------------------------------------------------------
// Pass 2: reduce 64 chunk-partials per sample, emit mean & 1/(std+eps).
// One wave per sample (8 waves in one block).
// ---------------------------------------------------------------------------
__global__ __launch_bounds__(256) void zmln_finalize(const float* __restrict__ partials,
                                                     float* __restrict__ stats)
{
    const int tid  = threadIdx.x;
    const int b    = tid >> 5;
    const int lane = tid & 31;

    float s = 0.0f, q = 0.0f, c = 0.0f;
#pragma unroll
    for (int kk = 0; kk < kNCHUNK / 32; ++kk) {          // exactly 2 per lane
        const int k = lane + kk * 32;
        const float* p = partials + (size_t)(b * kNCHUNK + k) * 3;
        s += p[0]; q += p[1]; c += p[2];
    }
#pragma unroll
    for (int off = 16; off >= 1; off >>= 1) {
        s += __shfl_xor(s, off, 32);
        q += __shfl_xor(q, off, 32);
        c += __shfl_xor(c, off, 32);
    }
    if (lane == 0) {
        const bool  safe = c > 0.0f;
        const float mean = safe ? s / fmaxf(c, 1.0f) : 0.0f;
        const float var  = (q - c * mean * mean) / fmaxf(c - 1.0f, 1.0f); // unbiased
        const float sd   = safe ? sqrtf(fmaxf(var, 0.0f)) : 1.0f;
        stats[b]     = mean;
        stats[8 + b] = 1.0f / (sd + kEPS);
    }
}

// ---------------------------------------------------------------------------
// Pass 3: y = a*x + bias, a = gamma[c]*inv_b, bias = beta[c] - a*mean_b.
// Pure stream: NT b128 loads + NT b128 stores (write-once output).
// total4 / (grid*block) = 16 exactly -> constant-trip loop, no exec churn.
// ---------------------------------------------------------------------------
__global__ __launch_bounds__(256) void zmln_norm(const float* __restrict__ x,
                                                 const float* __restrict__ gamma,
                                                 const float* __restrict__ beta,
                                                 const float* __restrict__ stats,
                                                 float* __restrict__ y)
{
    constexpr size_t total4 = (size_t)kB * kC * kS / 4;      // 16,777,216
    constexpr size_t stride = (size_t)kNBLK3 * 256;          // 1,048,576
    constexpr int    iters  = (int)(total4 / stride);        // 16

    const size_t base = (size_t)blockIdx.x * 256 + threadIdx.x;
    const f4* xf4 = (const f4*)x;
    f4*       yf4 = (f4*)y;

#pragma unroll 4
    for (int k = 0; k < iters; ++k) {
        const size_t i = base + (size_t)k * stride;
        const size_t e = i * 4;
        const int b = (int)(e >> 23);        // C*S = 2^23
        const int c = (int)(e >> 18) & 31;   // S   = 2^18
        const float mean = stats[b];
        const float inv  = stats[8 + b];
        const float a    = gamma[c] * inv;
        const float bias = beta[c] - a * mean;

        f4 v = __builtin_nontemporal_load(xf4 + i);
        f4 r;
        r.x = fmaf(a, v.x, bias);
        r.y = fmaf(a, v.y, bias);
        r.z = fmaf(a, v.z, bias);
        r.w = fmaf(a, v.w, bias);
        __builtin_nontemporal_store(r, yf4 + i);
    }
}

// ---------------------------------------------------------------------------
extern "C" void kernel_launch(void* const* d_in, const int* in_sizes, int n_in,
                              void* d_out, int out_size, void* d_ws, size_t ws_size,
                              hipStream_t stream) {
    const float* x     = (const float*)d_in[0];
    const float* gamma = (const float*)d_in[1];
    const float* beta  = (const float*)d_in[2];
    float*       y     = (float*)d_out;
    float*       ws    = (float*)d_ws;

    float* partials = ws;                 // 512 * 3 floats
    float* stats    = ws + kNBLK1 * 3;    // mean[8], inv[8]

    zmln_reduce  <<<kNBLK1, 256, 0, stream>>>(x, partials);
    zmln_finalize<<<1,      256, 0, stream>>>(partials, stats);
    zmln_norm    <<<kNBLK3, 256, 0, stream>>>(x, gamma, beta, stats, y);
}